// wFMLayer_1039382086093
// MI455X (gfx1250) — compile-verified
//
#include <hip/hip_runtime.h>
#include <hip/hip_bf16.h>
#include <math.h>

typedef float v2f __attribute__((ext_vector_type(2)));
typedef float v8f __attribute__((ext_vector_type(8)));

#define BN   4
#define NP   4096
#define DD   4
#define CC   8
#define KNN  16
#define COUT 16
#define FW   32          // D*C feature width
#define WPW  4           // waves per workgroup
#define TPB  (WPW * 32)
#define NLIST (WPW * 2)  // partial top-k lists per query

// ---------------- Kernel 1: per-point squared norms ----------------
__global__ __launch_bounds__(256) void k_norms(const float* __restrict__ flat,
                                               float* __restrict__ sq) {
    int p = blockIdx.x * blockDim.x + threadIdx.x;   // 0 .. B*N-1
    if (p >= BN * NP) return;
    const float* r = flat + (size_t)p * FW;
    float s = 0.f;
    #pragma unroll
    for (int f = 0; f < FW; ++f) s += r[f] * r[f];
    sq[p] = s;
}

// ---------------- Kernel 2: fused WMMA distance tiles + top-16 ----------------
// 4 waves per workgroup, each wave owns 1/4 of the candidate tiles and a private
// LDS key tile. All 32 lanes participate in selection (8 columns each). Partial
// sorted lists are merged stably (key, then index) at the end.
__global__ __launch_bounds__(TPB) void k_knn(const float* __restrict__ flat,
                                             const float* __restrict__ sq,
                                             int* __restrict__ idxout) {
    __shared__ float Kt[WPW][16][17];        // per-wave key tile (padded rows)
    __shared__ float Ld[NLIST][16][KNN];     // partial list keys   [list][query][slot]
    __shared__ int   Li[NLIST][16][KNN];     // partial list indices

    const int wg = blockIdx.x;               // 0 .. B*N/16-1
    const int b  = wg / (NP / 16);
    const int m0 = (wg % (NP / 16)) * 16;
    const float* fb  = flat + (size_t)b * NP * FW;
    const float* sqb = sq + (size_t)b * NP;

    const int tid  = threadIdx.x;
    const int w    = tid >> 5;               // wave id 0..3
    const int lane = tid & 31;
    const int l    = lane & 15;              // query row / tile column
    const int hf   = lane >> 4;              // K-half (A/B frags) and column-half (scan)

    // A fragments (16x4 fp32 layout): lane<16 -> K = 4kk+{0,1}, lane>=16 -> K = 4kk+{2,3}
    v2f a[8];
    #pragma unroll
    for (int kk = 0; kk < 8; ++kk)
        a[kk] = *(const v2f*)(fb + (size_t)(m0 + l) * FW + 4 * kk + 2 * hf);

    float topd[KNN];
    int   topi[KNN];
    #pragma unroll
    for (int s = 0; s < KNN; ++s) { topd[s] = __builtin_inff(); topi[s] = 0; }

    for (int t = w; t < NP / 16; t += WPW) {
        const int n0 = t * 16;
        // B fragments: B[k][n] = flat[n][k] -> same per-lane gather as A, candidate rows
        v2f bf[8];
        #pragma unroll
        for (int kk = 0; kk < 8; ++kk)
            bf[kk] = *(const v2f*)(fb + (size_t)(n0 + l) * FW + 4 * kk + 2 * hf);

        v8f c = {0.f, 0.f, 0.f, 0.f, 0.f, 0.f, 0.f, 0.f};
        #pragma unroll
        for (int kk = 0; kk < 8; ++kk)
            c = __builtin_amdgcn_wmma_f32_16x16x4_f32(
                    /*neg_a=*/false, a[kk], /*neg_b=*/false, bf[kk],
                    /*c_mod=*/(short)0, c, /*reuse_a=*/false, /*reuse_b=*/false);

        // Column index is fixed per lane -> fold candidate norm in before transpose.
        // key = sq_n - 2*G  (per-query constant sq_q dropped: ranking unchanged)
        const float sqn = sqb[n0 + l];
        #pragma unroll
        for (int r = 0; r < 8; ++r)
            Kt[w][r + 8 * hf][l] = fmaf(-2.0f, c[r], sqn);

        // same-wave LDS RAW: DS pipeline is in-order per wave; wait + stop reordering
        asm volatile("s_wait_dscnt 0x0" ::: "memory");

        // every lane scans 8 columns of query row l
        #pragma unroll
        for (int jj = 0; jj < 8; ++jj) {
            const int j = hf * 8 + jj;
            float key = Kt[w][l][j];
            int   ni  = n0 + j;
            if (key < topd[KNN - 1]) {
                #pragma unroll
                for (int s = 0; s < KNN; ++s) {
                    bool  mv = key < topd[s];   // strict: stable, lower index first
                    float td = topd[s];
                    int   ti = topi[s];
                    topd[s] = mv ? key : td;
                    topi[s] = mv ? ni  : ti;
                    key     = mv ? td  : key;
                    ni      = mv ? ti  : ni;
                }
            }
        }
    }

    // dump the 8 partial lists (per query) and merge
    const int listid = w * 2 + hf;
    #pragma unroll
    for (int s = 0; s < KNN; ++s) {
        Ld[listid][l][s] = topd[s];
        Li[listid][l][s] = topi[s];
    }
    __syncthreads();

    if (tid < 16) {
        const int q = tid;
        int head[NLIST];
        #pragma unroll
        for (int p = 0; p < NLIST; ++p) head[p] = 0;
        int* o = idxout + ((size_t)b * NP + m0 + q) * KNN;
        #pragma unroll 1
        for (int s = 0; s < KNN; ++s) {
            float bk = __builtin_inff();
            int   bi = 0x7fffffff;
            int   bl = 0;
            #pragma unroll
            for (int p = 0; p < NLIST; ++p) {
                const int h = head[p];
                if (h < KNN) {
                    const float kv = Ld[p][q][h];
                    const int   iv = Li[p][q][h];
                    const bool better = (kv < bk) || (kv == bk && iv < bi);
                    if (better) { bk = kv; bi = iv; bl = p; }
                }
            }
            o[s] = bi;
            #pragma unroll
            for (int p = 0; p < NLIST; ++p) head[p] += (p == bl) ? 1 : 0;
        }
    }
}

// ---------------- Kernel 3: gather + weighted sum + channel mix ----------------
__global__ __launch_bounds__(256) void k_out(const float* __restrict__ flat,
                                             const float* __restrict__ w1,
                                             const float* __restrict__ w2,
                                             const int* __restrict__ idx,
                                             float* __restrict__ out) {
    __shared__ float w1n[CC][KNN];    // row-normalized (dim=1)
    __shared__ float w2n[CC][COUT];   // col-normalized (dim=0)
    int t = threadIdx.x;
    if (t < CC) {
        float s = 0.f;
        #pragma unroll
        for (int k = 0; k < KNN; ++k) { float v = w1[t * KNN + k]; s += v * v; }
        float inv = 1.0f / sqrtf(s);
        #pragma unroll
        for (int k = 0; k < KNN; ++k) w1n[t][k] = w1[t * KNN + k] * inv;
    } else if (t < CC + COUT) {
        int o = t - CC;
        float s = 0.f;
        #pragma unroll
        for (int c = 0; c < CC; ++c) { float v = w2[c * COUT + o]; s += v * v; }
        float inv = 1.0f / sqrtf(s);
        #pragma unroll
        for (int c = 0; c < CC; ++c) w2n[c][o] = w2[c * COUT + o] * inv;
    }
    __syncthreads();

    int tid = blockIdx.x * blockDim.x + t;     // over B*N*D
    if (tid >= BN * NP * DD) return;
    int d  = tid % DD;
    int p  = tid / DD;                         // b*N + n
    int bb = p / NP;

    const int* ip = idx + (size_t)p * KNN;
    float acc[CC];
    #pragma unroll
    for (int c = 0; c < CC; ++c) acc[c] = 0.f;

    for (int k = 0; k < KNN; ++k) {
        int nb = ip[k];
        const float* fr = flat + ((size_t)bb * NP + nb) * FW + d * CC;
        #pragma unroll
        for (int c = 0; c < CC; ++c) acc[c] += fr[c] * w1n[c][k];
    }

    float* op = out + (size_t)tid * COUT;
    #pragma unroll
    for (int o = 0; o < COUT; ++o) {
        float s = 0.f;
        #pragma unroll
        for (int c = 0; c < CC; ++c) s += acc[c] * w2n[c][o];
        op[o] = s;
    }
}

// ---------------- Launch ----------------
extern "C" void kernel_launch(void* const* d_in, const int* in_sizes, int n_in,
                              void* d_out, int out_size, void* d_ws, size_t ws_size,
                              hipStream_t stream) {
    const float* x  = (const float*)d_in[0];   // [B,N,D,C] == flat [B,N,32]
    const float* w1 = (const float*)d_in[1];   // [C,K]
    const float* w2 = (const float*)d_in[2];   // [C,Cout]
    float* out = (float*)d_out;                // [B,N,D,Cout]

    float* sq  = (float*)d_ws;                                          // B*N floats
    int*   idx = (int*)((char*)d_ws + (size_t)BN * NP * sizeof(float)); // B*N*K ints

    k_norms<<<(BN * NP + 255) / 256, 256, 0, stream>>>(x, sq);
    k_knn<<<BN * NP / 16, TPB, 0, stream>>>(x, sq, idx);
    k_out<<<(BN * NP * DD + 255) / 256, 256, 0, stream>>>(x, w1, w2, idx, out);
}